// Model_18176301597001
// MI455X (gfx1250) — compile-verified
//
#include <hip/hip_runtime.h>
#include <math.h>

typedef __attribute__((ext_vector_type(2))) float v2f;
typedef __attribute__((ext_vector_type(8))) float v8f;

__device__ __forceinline__ float fast_tanh(float x) {
#if __has_builtin(__builtin_amdgcn_tanhf)
    return __builtin_amdgcn_tanhf(x);      // gfx1250 v_tanh_f32 (trans pipe)
#else
    return tanhf(x);
#endif
}

__device__ __forceinline__ float fast_sigmoid(float x) {
    // 1/(1+exp(-x)) = rcp(1 + exp2(-x * log2(e)))
    float e = __builtin_amdgcn_exp2f(-1.44269504088896340736f * x);
    return __builtin_amdgcn_rcpf(1.0f + e);
}

// out[m] = sigmoid( W2 . tanh(W1 @ [r_m, theta_m] + b1) + b2 )
//
// One wave32 handles 32 points per iteration (two 16-row WMMA groups):
//   - every lane computes the polar transform of its own point
//   - group 0: A rows = points base+0..15  (r,theta already in lanes 0-15)
//   - group 1: A rows = points base+16..31 (r,theta moved down via shfl_xor 16)
//   A  (16x4 f32): M = lane&15, K0=r, K1=theta, K2=K3=0 (half1 of A zeroed)
//   B  (4x16 f32): N = 16t+(lane&15); half0: K0=W1[n][0], K1=W1[n][1]; half1 (K2,K3)=0
//   C/D (16x16 f32): element r of lane l is (M = r + 8*(l>=16), N = l&15), seeded b1[N]
__global__ __launch_bounds__(256) void mlp_polar_wmma(
    const float* __restrict__ input, const float* __restrict__ W1,
    const float* __restrict__ b1,    const float* __restrict__ W2,
    const float* __restrict__ b2,    float* __restrict__ out, int batch)
{
    const int lane   = threadIdx.x & 31;
    const int n0     = lane & 15;      // hidden-unit column within a 16-wide tile
    const bool half0 = lane < 16;

    const int wavesPerBlock = blockDim.x >> 5;
    const int waveId   = blockIdx.x * wavesPerBlock + (threadIdx.x >> 5);
    const int numWaves = gridDim.x * wavesPerBlock;

    // Loop-invariant per-lane weights for the 4 hidden tiles
    v2f   bm[4];
    float b1v[4], w2v[4];
#pragma unroll
    for (int t = 0; t < 4; ++t) {
        const int n = t * 16 + n0;
        bm[t].x = half0 ? W1[2 * n + 0] : 0.0f;   // B K=0 row (half1 = K=2 row = 0)
        bm[t].y = half0 ? W1[2 * n + 1] : 0.0f;   // B K=1 row (half1 = K=3 row = 0)
        b1v[t]  = b1[n];
        w2v[t]  = W2[n];
    }
    const float b2v = b2[0];

    const int numGroups = batch >> 5;              // 32 points per group
    for (int g = waveId; g < numGroups; g += numWaves) {
        const int base = g << 5;

        // Each lane loads + transforms its own point (fully coalesced 256B)
        const float2 xy = ((const float2*)input)[base + lane];
        const float r  = sqrtf(xy.x * xy.x + xy.y * xy.y);
        const float th = atan2f(xy.y, xy.x);

        // Hand points 16..31 down into lanes 0..15 for the second A matrix
        const float rB  = __shfl_xor(r,  16, 32);
        const float thB = __shfl_xor(th, 16, 32);

        v2f a0, a1;
        a0.x = half0 ? r   : 0.0f;   a0.y = half0 ? th  : 0.0f;
        a1.x = half0 ? rB  : 0.0f;   a1.y = half0 ? thB : 0.0f;

        // Distinct accumulators -> 8 independent WMMAs, no WAR serialization
        v8f c0[4], c1[4];
#pragma unroll
        for (int t = 0; t < 4; ++t) {
#pragma unroll
            for (int i = 0; i < 8; ++i) { c0[t][i] = b1v[t]; c1[t][i] = b1v[t]; }
        }
#pragma unroll
        for (int t = 0; t < 4; ++t) {
            c0[t] = __builtin_amdgcn_wmma_f32_16x16x4_f32(
                        false, a0, false, bm[t], (short)0, c0[t], false, false);
            c1[t] = __builtin_amdgcn_wmma_f32_16x16x4_f32(
                        false, a1, false, bm[t], (short)0, c1[t], false, false);
        }

        // Layer 2: tanh then scale by W2[n]; 64 independent trans chains
        v8f p0 = {}, p1 = {};
#pragma unroll
        for (int t = 0; t < 4; ++t) {
#pragma unroll
            for (int i = 0; i < 8; ++i) {
                p0[i] += fast_tanh(c0[t][i]) * w2v[t];
                p1[i] += fast_tanh(c1[t][i]) * w2v[t];
            }
        }

        // Reduce across hidden units: butterfly within each 16-lane half
        // (masks < 16 keep the halves independent, matching the row split)
        float res0[8], res1[8];
#pragma unroll
        for (int i = 0; i < 8; ++i) {
            float s0 = p0[i], s1 = p1[i];
            s0 += __shfl_xor(s0, 1, 32);  s1 += __shfl_xor(s1, 1, 32);
            s0 += __shfl_xor(s0, 2, 32);  s1 += __shfl_xor(s1, 2, 32);
            s0 += __shfl_xor(s0, 4, 32);  s1 += __shfl_xor(s1, 4, 32);
            s0 += __shfl_xor(s0, 8, 32);  s1 += __shfl_xor(s1, 8, 32);
            res0[i] = fast_sigmoid(s0 + b2v);
            res1[i] = fast_sigmoid(s1 + b2v);
        }

        // Group 0: lane 0 -> rows 0..7, lane 16 -> rows 8..15
        // Group 1: same but offset by 16 points
        if (n0 == 0) {
            float4* o0 = (float4*)(out + base + (half0 ? 0 : 8));
            o0[0] = make_float4(res0[0], res0[1], res0[2], res0[3]);
            o0[1] = make_float4(res0[4], res0[5], res0[6], res0[7]);
            float4* o1 = (float4*)(out + base + 16 + (half0 ? 0 : 8));
            o1[0] = make_float4(res1[0], res1[1], res1[2], res1[3]);
            o1[1] = make_float4(res1[4], res1[5], res1[6], res1[7]);
        }
    }
}

extern "C" void kernel_launch(void* const* d_in, const int* in_sizes, int n_in,
                              void* d_out, int out_size, void* d_ws, size_t ws_size,
                              hipStream_t stream) {
    const float* input = (const float*)d_in[0];
    const float* W1    = (const float*)d_in[1];
    const float* b1    = (const float*)d_in[2];
    const float* W2    = (const float*)d_in[3];
    const float* b2    = (const float*)d_in[4];
    float* out = (float*)d_out;

    const int batch = in_sizes[0] / 2;      // input is (BATCH, 2); 2^21 -> /32 exact
    const int threads = 256;                // 8 wave32s per block
    const int blocks  = 2048;               // 16384 waves -> 4 groups of 32 pts each

    mlp_polar_wmma<<<blocks, threads, 0, stream>>>(input, W1, b1, W2, b2, out, batch);
}